// ASAP_Pooling_20658792694517
// MI455X (gfx1250) — compile-verified
//
#include <hip/hip_runtime.h>
#include <hip/hip_bf16.h>
#include <cstdint>

// ---------------- problem constants ----------------
static constexpr int B    = 16;
static constexpr int NPG  = 512;
static constexpr int EG   = 8192;
static constexpr int C    = 128;
static constexpr int H    = 2;
static constexpr int K    = 128;
static constexpr int N    = B * NPG;          // 8192 nodes total
static constexpr int BEG  = B * EG;           // 131072 input edges
static constexpr int ETOT = BEG + N;          // + self loops = 139264

typedef float v2f __attribute__((ext_vector_type(2)));
typedef float v8f __attribute__((ext_vector_type(8)));

// ---------------- WMMA helper (f32 16x16x4, CDNA5) ----------------
__device__ __forceinline__ v8f wmma4(v2f a, v2f b, v8f c) {
#if defined(__gfx1250__) && __has_builtin(__builtin_amdgcn_wmma_f32_16x16x4_f32)
  // (neg_a, A, neg_b, B, c_mod, C, reuse_a, reuse_b)
  return __builtin_amdgcn_wmma_f32_16x16x4_f32(false, a, false, b, (short)0, c,
                                               false, false);
#else
  // host-pass placeholder (never taken on gfx1250 device pass)
  c[0] += a[0] * b[0] + a[1] * b[1];
  return c;
#endif
}

// async-stage a 64KB row-major [128,128] f32 matrix into LDS (whole block).
// CDNA5: GLOBAL_LOAD_ASYNC_TO_LDS_B128 (ASYNCcnt) + s_wait_asynccnt, per
// CDNA5 ISA. Low 32 bits of a generic shared pointer
// are the LDS byte offset (aperture rule, CDNA5 ISA).
__device__ __forceinline__ void stage_B_to_lds(float* Bs, const float* __restrict__ Bm) {
#if defined(__gfx1250__)
  for (int i = threadIdx.x; i < (C * C) / 4; i += blockDim.x) {
    unsigned lds_off = (unsigned)(uintptr_t)(&Bs[i * 4]);
    const void* gp = (const void*)(Bm + i * 4);
    asm volatile("global_load_async_to_lds_b128 %0, %1, off"
                 :: "v"(lds_off), "v"(gp) : "memory");
  }
  asm volatile("s_wait_asynccnt 0x0" ::: "memory");
#else
  for (int i = threadIdx.x; i < C * C; i += blockDim.x) Bs[i] = Bm[i];
#endif
  __syncthreads();
}

// ---------------- utility ----------------
__global__ void fill_f32(float* p, float v, int n) {
  int i = blockIdx.x * blockDim.x + threadIdx.x;
  if (i < n) p[i] = v;
}

__device__ __forceinline__ float sigmoidf_(float x) {
  return 1.0f / (1.0f + __expf(-x));
}

// encode float -> monotonic unsigned (for atomicMax-based segment max)
__device__ __forceinline__ unsigned ord_enc(float f) {
  unsigned u = __float_as_uint(f);
  return (u & 0x80000000u) ? ~u : (u | 0x80000000u);
}
__device__ __forceinline__ float ord_dec(unsigned u) {
  u = (u & 0x80000000u) ? (u & 0x7fffffffu) : ~u;
  return __uint_as_float(u);
}

// ---------------- GEMM: D[M,128] = A[M,128] @ Bm[128,128] (+bias) ----------
// Bm staged to LDS once per block (async-to-LDS), one wave per 16x16 output
// tile, WMMA f32 16x16x4, K swept 128/4 = 32 steps. Grid must be exactly
// M/8 blocks of 128 threads (4 waves = 4 consecutive tiles).
__global__ void gemm128_bias(const float* __restrict__ A,
                             const float* __restrict__ Bm,
                             const float* __restrict__ bias,
                             float* __restrict__ D) {
  __shared__ float Bs[C * C];                 // 64 KB
  stage_B_to_lds(Bs, Bm);

  int lane = threadIdx.x & 31;
  int wave = threadIdx.x >> 5;
  int t = blockIdx.x * 4 + wave;
  int m0 = (t >> 3) << 4;
  int n0 = (t & 7) << 4;
  int mrow  = m0 + (lane & 15);               // A-matrix: lanes 0-15 / 16-31 hold M=0..15
  int ncol  = n0 + (lane & 15);               // B-matrix: N striped across lanes
  int khalf = (lane >> 4) << 1;               // lanes 16-31 carry K+2,K+3
  v8f acc = {};
  for (int kk = 0; kk < C; kk += 4) {
    v2f a, b;
    a[0] = A[(size_t)mrow * C + kk + khalf];
    a[1] = A[(size_t)mrow * C + kk + khalf + 1];
    b[0] = Bs[(kk + khalf) * C + ncol];       // ds_load from staged weights
    b[1] = Bs[(kk + khalf + 1) * C + ncol];
    acc = wmma4(a, b, acc);
  }
  float bv = bias ? bias[ncol] : 0.0f;
  int rbase = m0 + (lane >> 4) * 8;           // C/D: vgpr i -> M = i (+8 for hi lanes)
  for (int i = 0; i < 8; ++i)
    D[(size_t)(rbase + i) * C + ncol] = acc[i] + bv;
}

// ---------------- Em = S^T (A S) per graph, diag forced to 1 ---------------
// S, AS: [B, NPG, K]; Em: [B, K, K]. A-operand read transposed from S.
// (B-panel is 256KB per graph -> stays on the global path.)
__global__ void gemm_tn_em(const float* __restrict__ S,
                           const float* __restrict__ AS,
                           float* __restrict__ Em) {
  int lane = threadIdx.x & 31;
  int wave = threadIdx.x >> 5;
  int t = blockIdx.x * 4 + wave;              // 16 graphs * 64 tiles = 1024
  int g  = t >> 6;
  int tt = t & 63;
  int m0 = (tt >> 3) << 4;
  int n0 = (tt & 7) << 4;
  const float* Sg  = S  + (size_t)g * NPG * K;
  const float* ASg = AS + (size_t)g * NPG * K;
  int mi = m0 + (lane & 15);
  int ni = n0 + (lane & 15);
  int khalf = (lane >> 4) << 1;
  v8f acc = {};
  for (int kk = 0; kk < NPG; kk += 4) {
    v2f a, b;
    a[0] = Sg[(size_t)(kk + khalf) * K + mi];       // A[m,n-dim] = S[n, m]
    a[1] = Sg[(size_t)(kk + khalf + 1) * K + mi];
    b[0] = ASg[(size_t)(kk + khalf) * K + ni];
    b[1] = ASg[(size_t)(kk + khalf + 1) * K + ni];
    acc = wmma4(a, b, acc);
  }
  int rbase = m0 + (lane >> 4) * 8;
  for (int i = 0; i < 8; ++i) {
    int r = rbase + i;
    float v = (r == ni) ? 1.0f : acc[i];            // Em*(1-eye)+eye
    Em[((size_t)g * K + r) * K + ni] = v;
  }
}

// ---------------- edge list with self loops (global node ids) -------------
__global__ void build_edges(const int* __restrict__ ei,
                            int* __restrict__ srcf, int* __restrict__ dstf) {
  int i = blockIdx.x * blockDim.x + threadIdx.x;
  if (i >= ETOT) return;
  if (i < BEG) { srcf[i] = ei[i]; dstf[i] = ei[BEG + i]; }
  else { int n = i - BEG; srcf[i] = n; dstf[i] = n; }
}

// ---------------- attention projections a_src.h, a_dst.h ------------------
__global__ void attn_coef(const float* __restrict__ h,
                          const float* __restrict__ a_src,
                          const float* __restrict__ a_dst,
                          float* __restrict__ al, float* __restrict__ ar) {
  int i = blockIdx.x * blockDim.x + threadIdx.x;    // N*H
  if (i >= N * H) return;
  int n = i >> 1, hh = i & 1;
  const float* hp = h + (size_t)n * C + hh * (C / H);
  float s1 = 0.f, s2 = 0.f;
  for (int c = 0; c < C / H; ++c) {
    float v = hp[c];
    s1 += v * a_src[hh * (C / H) + c];
    s2 += v * a_dst[hh * (C / H) + c];
  }
  al[i] = s1; ar[i] = s2;
}

__global__ void edge_max(const float* __restrict__ al, const float* __restrict__ ar,
                         const int* __restrict__ srcf, const int* __restrict__ dstf,
                         unsigned* __restrict__ mbuf) {
  int i = blockIdx.x * blockDim.x + threadIdx.x;    // ETOT*H
  if (i >= ETOT * H) return;
  int e = i >> 1, hh = i & 1;
  int s = srcf[e], d = dstf[e];
  float x = al[s * H + hh] + ar[d * H + hh];
  x = (x > 0.f) ? x : 0.2f * x;                     // leaky_relu(0.2)
  atomicMax(&mbuf[s * H + hh], ord_enc(x));
}

__global__ void edge_exp(const float* __restrict__ al, const float* __restrict__ ar,
                         const int* __restrict__ srcf, const int* __restrict__ dstf,
                         const unsigned* __restrict__ mbuf,
                         float* __restrict__ wbuf, float* __restrict__ denom) {
  int i = blockIdx.x * blockDim.x + threadIdx.x;    // ETOT*H
  if (i >= ETOT * H) return;
  int e = i >> 1, hh = i & 1;
  int s = srcf[e], d = dstf[e];
  float x = al[s * H + hh] + ar[d * H + hh];
  x = (x > 0.f) ? x : 0.2f * x;
  float w = __expf(x - ord_dec(mbuf[s * H + hh]));
  wbuf[i] = w;
  atomicAdd(&denom[s * H + hh], w);
}

// out[src] += alpha * h[dst]; score = mean_h alpha
__global__ void edge_agg1(const float* __restrict__ h, const float* __restrict__ wbuf,
                          const float* __restrict__ denom,
                          const int* __restrict__ srcf, const int* __restrict__ dstf,
                          float* __restrict__ outb, float* __restrict__ score) {
  int e = blockIdx.x;
  int c = threadIdx.x;
  int s = srcf[e], d = dstf[e];
  float a0 = wbuf[e * H + 0] / denom[s * H + 0];
  float a1 = wbuf[e * H + 1] / denom[s * H + 1];
  float alpha = (c < C / H) ? a0 : a1;
  atomicAdd(&outb[(size_t)s * C + c], alpha * h[(size_t)d * C + c]);
  if (c == 0) score[e] = 0.5f * (a0 + a1);
}

// out2[src] += score * out[dst]
__global__ void edge_agg2(const float* __restrict__ outb, const float* __restrict__ score,
                          const int* __restrict__ srcf, const int* __restrict__ dstf,
                          float* __restrict__ out2) {
  int e = blockIdx.x;
  int c = threadIdx.x;
  int s = srcf[e], d = dstf[e];
  atomicAdd(&out2[(size_t)s * C + c], outb[(size_t)d * C + c] * score[e]);
}

__global__ void leconv_dots(const float* __restrict__ out2,
                            const float* __restrict__ W1, const float* __restrict__ b1,
                            const float* __restrict__ W2, const float* __restrict__ W3,
                            const float* __restrict__ b3,
                            float* __restrict__ p1, float* __restrict__ p2,
                            float* __restrict__ p3) {
  int n = blockIdx.x * blockDim.x + threadIdx.x;
  if (n >= N) return;
  float s1 = 0.f, s2 = 0.f, s3 = 0.f;
  const float* o2 = out2 + (size_t)n * C;
  for (int c = 0; c < C; ++c) {
    float v = o2[c];
    s1 += v * W1[c]; s2 += v * W2[c]; s3 += v * W3[c];
  }
  p1[n] = s1 + b1[0]; p2[n] = s2; p3[n] = s3 + b3[0];
}

__global__ void edge_le(const int* __restrict__ srcf, const int* __restrict__ dstf,
                        const float* __restrict__ p1, const float* __restrict__ p2,
                        float* __restrict__ agg) {
  int e = blockIdx.x * blockDim.x + threadIdx.x;
  if (e >= ETOT) return;
  int s = srcf[e], d = dstf[e];
  atomicAdd(&agg[d], p1[s] - p2[d]);
}

__global__ void pool_x(const float* __restrict__ x, float* __restrict__ xro) {
  int i = blockIdx.x * blockDim.x + threadIdx.x;    // B*C
  if (i >= B * C) return;
  int g = i / C, c = i % C;
  float s = 0.f;
  for (int n = 0; n < NPG; ++n) s += x[(size_t)(g * NPG + n) * C + c];
  xro[i] = s;
}

__global__ void readout_vec(const float* __restrict__ xro, const float* __restrict__ Wr,
                            const float* __restrict__ br, float* __restrict__ rv) {
  int i = blockIdx.x * blockDim.x + threadIdx.x;    // B*C
  if (i >= B * C) return;
  int g = i / C, c = i % C;
  float s = 0.f;
  for (int k = 0; k < C; ++k) s += xro[g * C + k] * Wr[(size_t)k * C + c];
  rv[i] = s + br[c];
}

__global__ void fitness_k(const float* __restrict__ rv, const float* __restrict__ og,
                          const float* __restrict__ agg, const float* __restrict__ p3,
                          float* __restrict__ fit) {
  int n = blockIdx.x * blockDim.x + threadIdx.x;
  if (n >= N) return;
  int g = n >> 9;                                   // / NPG
  float s = 0.f;
  for (int c = 0; c < C; ++c) s += rv[g * C + c] * og[(size_t)n * C + c];
  fit[n] = sigmoidf_(agg[n] + p3[n]) + sigmoidf_(s);
}

// top-K per graph by rank counting; reproduces lax.top_k ordering/ties
__global__ void topk_k(const float* __restrict__ fit, int* __restrict__ pidx,
                       float* __restrict__ fitk, int* __restrict__ inv,
                       int* __restrict__ sel, float* __restrict__ batcho,
                       float* __restrict__ permo) {
  __shared__ float sf[NPG];
  int g = blockIdx.x;
  int i = threadIdx.x;
  float f = fit[g * NPG + i];
  sf[i] = f;
  __syncthreads();
  int rank = 0;
  for (int j = 0; j < NPG; ++j) {
    float fj = sf[j];
    rank += (fj > f) || (fj == f && j < i);
  }
  if (rank < K) {
    pidx[g * K + rank]  = i;
    fitk[g * K + rank]  = f;
    inv[g * NPG + i]    = rank;
    sel[g * NPG + i]    = 1;
    batcho[g * K + rank] = (float)g;
    permo[g * K + rank]  = (float)(g * NPG + i);
  }
}

__global__ void gather_x(const float* __restrict__ out2, const int* __restrict__ pidx,
                         const float* __restrict__ fitk, float* __restrict__ Xo) {
  int i = blockIdx.x * blockDim.x + threadIdx.x;    // B*K*C
  if (i >= B * K * C) return;
  int c = i & (C - 1);
  int gk = i >> 7;
  int g = gk >> 7;
  Xo[i] = out2[(size_t)(g * NPG + pidx[gk]) * C + c] * fitk[gk];
}

// S[dst, inv[src]] += score for selected src (S laid out [B*NPG, K])
__global__ void s_fill(const int* __restrict__ srcf, const int* __restrict__ dstf,
                       const int* __restrict__ sel, const int* __restrict__ inv,
                       const float* __restrict__ score, float* __restrict__ So) {
  int e = blockIdx.x * blockDim.x + threadIdx.x;
  if (e >= ETOT) return;
  int s = srcf[e];
  if (sel[s]) atomicAdd(&So[(size_t)dstf[e] * K + inv[s]], score[e]);
}

// AS[src, :] += S[dst, :]  (A has a unit entry per edge, dups accumulate)
__global__ void as_k(const int* __restrict__ srcf, const int* __restrict__ dstf,
                     const float* __restrict__ So, float* __restrict__ AS) {
  int e = blockIdx.x;
  int k = threadIdx.x;
  atomicAdd(&AS[(size_t)srcf[e] * K + k], So[(size_t)dstf[e] * K + k]);
}

// ew[g,i,j,c] = (xk[g,i,c] + xk[g,j,c]) * 0.5 * Em[g,i,j]
__global__ void ew_k(const float* __restrict__ xk, const float* __restrict__ Em,
                     float* __restrict__ ewo) {
  size_t idx = blockIdx.x;                          // g*K*K + i*K + j
  int c = threadIdx.x;
  int j  = (int)(idx & (K - 1));
  int gi = (int)(idx >> 7);
  int i  = gi & (K - 1);
  int g  = gi >> 7;
  float em = Em[idx];
  ewo[idx * C + c] =
      (xk[(size_t)(g * K + i) * C + c] + xk[(size_t)(g * K + j) * C + c]) * 0.5f * em;
}

// ---------------- host driver ----------------
extern "C" void kernel_launch(void* const* d_in, const int* in_sizes, int n_in,
                              void* d_out, int out_size, void* d_ws, size_t ws_size,
                              hipStream_t stream) {
  const float* x     = (const float*)d_in[0];
  const int*   ei    = (const int*)d_in[1];
  const float* Wg    = (const float*)d_in[3];
  const float* a_src = (const float*)d_in[4];
  const float* a_dst = (const float*)d_in[5];
  const float* W1    = (const float*)d_in[6];
  const float* b1    = (const float*)d_in[7];
  const float* W2    = (const float*)d_in[8];
  const float* W3    = (const float*)d_in[9];
  const float* b3    = (const float*)d_in[10];
  const float* Wr    = (const float*)d_in[11];
  const float* br    = (const float*)d_in[12];
  const float* Wo    = (const float*)d_in[13];
  const float* bo    = (const float*)d_in[14];
  const float* Wx    = (const float*)d_in[15];
  const float* bx    = (const float*)d_in[16];

  // outputs, concatenated in reference return order
  float* Xo     = (float*)d_out;                 // [B,K,C]
  float* Emo    = Xo + (size_t)B * K * C;        // [B,K,K]
  float* batcho = Emo + (size_t)B * K * K;       // [B*K]
  float* permo  = batcho + (size_t)B * K;        // [B*K]
  float* fito   = permo + (size_t)B * K;         // [N]
  float* So     = fito + (size_t)N;              // [B,NPG,K]
  float* ewo    = So + (size_t)N * K;            // [B,K,K,C]

  // workspace
  float* ws = (float*)d_ws;
  size_t off = 0;
  float* h     = ws + off; off += (size_t)N * C;
  float* outb  = ws + off; off += (size_t)N * C;
  float* out2  = ws + off; off += (size_t)N * C;
  float* og    = ws + off; off += (size_t)N * C;
  float* AS    = ws + off; off += (size_t)N * K;
  float* xk    = ws + off; off += (size_t)B * K * C;
  float* al    = ws + off; off += (size_t)N * H;
  float* ar    = ws + off; off += (size_t)N * H;
  unsigned* mbuf = (unsigned*)(ws + off); off += (size_t)N * H;
  float* denom = ws + off; off += (size_t)N * H;
  float* wbuf  = ws + off; off += (size_t)ETOT * H;
  float* score = ws + off; off += (size_t)ETOT;
  float* p1    = ws + off; off += (size_t)N;
  float* p2    = ws + off; off += (size_t)N;
  float* p3    = ws + off; off += (size_t)N;
  float* agg   = ws + off; off += (size_t)N;
  float* xro   = ws + off; off += (size_t)B * C;
  float* rv    = ws + off; off += (size_t)B * C;
  float* fitk  = ws + off; off += (size_t)B * K;
  int* pidx = (int*)(ws + off); off += (size_t)B * K;
  int* inv  = (int*)(ws + off); off += (size_t)N;
  int* sel  = (int*)(ws + off); off += (size_t)N;
  int* srcf = (int*)(ws + off); off += (size_t)ETOT;
  int* dstf = (int*)(ws + off); off += (size_t)ETOT;

  auto cdiv = [](int a, int b) { return (a + b - 1) / b; };

  // zero/init accumulators every call (graph-replay deterministic)
  fill_f32<<<cdiv(N * C, 256), 256, 0, stream>>>(outb, 0.f, N * C);
  fill_f32<<<cdiv(N * C, 256), 256, 0, stream>>>(out2, 0.f, N * C);
  fill_f32<<<cdiv(N * K, 256), 256, 0, stream>>>(AS, 0.f, N * K);
  fill_f32<<<cdiv(N * H, 256), 256, 0, stream>>>(denom, 0.f, N * H);
  fill_f32<<<cdiv(N * H, 256), 256, 0, stream>>>((float*)mbuf, 0.f, N * H); // < enc(any finite)
  fill_f32<<<cdiv(N, 256), 256, 0, stream>>>(agg, 0.f, N);
  fill_f32<<<cdiv(N, 256), 256, 0, stream>>>((float*)sel, 0.f, N);
  fill_f32<<<cdiv(N, 256), 256, 0, stream>>>((float*)inv, 0.f, N);
  fill_f32<<<cdiv(N * K, 256), 256, 0, stream>>>(So, 0.f, N * K);

  build_edges<<<cdiv(ETOT, 256), 256, 0, stream>>>(ei, srcf, dstf);

  // h = x @ Wg           (WMMA, weights staged via async-to-LDS)
  gemm128_bias<<<N / 8, 128, 0, stream>>>(x, Wg, nullptr, h);

  attn_coef<<<cdiv(N * H, 256), 256, 0, stream>>>(h, a_src, a_dst, al, ar);
  edge_max<<<cdiv(ETOT * H, 256), 256, 0, stream>>>(al, ar, srcf, dstf, mbuf);
  edge_exp<<<cdiv(ETOT * H, 256), 256, 0, stream>>>(al, ar, srcf, dstf, mbuf, wbuf, denom);
  edge_agg1<<<ETOT, C, 0, stream>>>(h, wbuf, denom, srcf, dstf, outb, score);
  edge_agg2<<<ETOT, C, 0, stream>>>(outb, score, srcf, dstf, out2);

  leconv_dots<<<cdiv(N, 256), 256, 0, stream>>>(out2, W1, b1, W2, W3, b3, p1, p2, p3);
  edge_le<<<cdiv(ETOT, 256), 256, 0, stream>>>(srcf, dstf, p1, p2, agg);

  pool_x<<<cdiv(B * C, 256), 256, 0, stream>>>(x, xro);
  readout_vec<<<cdiv(B * C, 256), 256, 0, stream>>>(xro, Wr, br, rv);

  // o = out2 @ Wo + bo   (WMMA)
  gemm128_bias<<<N / 8, 128, 0, stream>>>(out2, Wo, bo, og);

  fitness_k<<<cdiv(N, 256), 256, 0, stream>>>(rv, og, agg, p3, fito);
  topk_k<<<B, NPG, 0, stream>>>(fito, pidx, fitk, inv, sel, batcho, permo);
  gather_x<<<cdiv(B * K * C, 256), 256, 0, stream>>>(out2, pidx, fitk, Xo);

  s_fill<<<cdiv(ETOT, 256), 256, 0, stream>>>(srcf, dstf, sel, inv, score, So);
  as_k<<<ETOT, K, 0, stream>>>(srcf, dstf, So, AS);

  // Em = S^T (A S), diag := 1   (WMMA, transposed A operand)
  gemm_tn_em<<<(B * 64) / 4, 128, 0, stream>>>(So, AS, Emo);

  // xk = X @ Wx + bx     (WMMA)
  gemm128_bias<<<(B * K) / 8, 128, 0, stream>>>(Xo, Wx, bx, xk);

  ew_k<<<B * K * K, C, 0, stream>>>(xk, Emo, ewo);
}